// DCNv2Pooling_1838246003415
// MI455X (gfx1250) — compile-verified
//
#include <hip/hip_runtime.h>

// Problem constants (match reference)
#define B_      4
#define C_      256
#define H_      128
#define W_      128
#define HW_     (H_ * W_)
#define NR_     512
#define P_      7
#define NBIN_   49          // P*P
#define NSAMP_  784         // NBIN_ * RATIO*RATIO
#define SCALE_  0.0625f
#define GAMMA_  0.1f
#define OUT_PER_ROI_ (C_ * NBIN_)   // 12544 floats

#define GLOBAL_AS __attribute__((address_space(1)))
#define LOCAL_AS  __attribute__((address_space(3)))

#if defined(__has_builtin)
#  if __has_builtin(__builtin_amdgcn_global_store_async_from_lds_b32) && \
      __has_builtin(__builtin_amdgcn_s_wait_asynccnt)
#    define HAVE_ASYNC_LDS 1
#  endif
#endif
#ifndef HAVE_ASYNC_LDS
#  define HAVE_ASYNC_LDS 0
#endif

// ---------------------------------------------------------------------------
// Compile-only probe: defined FIRST so the disasm snippet shows whether the
// gfx1250 async LDS->global path actually lowers (global_store_async_from_lds
// + s_wait_asynccnt). Never launched.
// ---------------------------------------------------------------------------
__global__ void cdna5_async_probe(float* g)
{
    __shared__ float buf[64];
    const int t = threadIdx.x;
    buf[t] = (float)t;
    __syncthreads();
#if HAVE_ASYNC_LDS
    __builtin_amdgcn_global_store_async_from_lds_b32(
        (GLOBAL_AS int*)(g + t), (LOCAL_AS int*)(&buf[t]), 0, 0);
    __builtin_amdgcn_s_wait_asynccnt(0);
#else
    g[t] = buf[t];
#endif
}

// ---------------------------------------------------------------------------
// Pass 1: (B, C, H*W) -> (B, H*W, C) tiled transpose so that channel gathers
// in the pooling kernel become unit-stride.
// ---------------------------------------------------------------------------
__global__ __launch_bounds__(256) void transpose_chw_hwc(
    const float* __restrict__ in, float* __restrict__ out)
{
    __shared__ float tile[32][33];   // +1 pad: no LDS bank conflicts
    const int b  = blockIdx.z;
    const int p0 = blockIdx.x * 32;  // pixel tile (H*W)
    const int c0 = blockIdx.y * 32;  // channel tile
    const int tx = threadIdx.x;      // 0..31
    const int ty = threadIdx.y;      // 0..7

    const float* inb  = in  + (size_t)b * C_ * HW_;
    float*       outb = out + (size_t)b * HW_ * C_;

#pragma unroll
    for (int i = 0; i < 32; i += 8)
        tile[ty + i][tx] = inb[(size_t)(c0 + ty + i) * HW_ + (p0 + tx)];
    __syncthreads();
#pragma unroll
    for (int i = 0; i < 32; i += 8)
        outb[(size_t)(p0 + ty + i) * C_ + (c0 + tx)] = tile[tx][ty + i];
}

__device__ __forceinline__ void fma4(float4& a, float w, const float4 v)
{
    a.x = fmaf(w, v.x, a.x);
    a.y = fmaf(w, v.y, a.y);
    a.z = fmaf(w, v.z, a.z);
    a.w = fmaf(w, v.w, a.w);
}

// ---------------------------------------------------------------------------
// Pass 2: one workgroup per ROI (256 threads = 8 waves).
//   Phase 1: build 784-entry (offset, weight) sample table in LDS.
//   Phase 2 (TR):   4 subgroups x 64 threads; each thread owns 4 channels and
//                   gathers float4 (16B/lane -> 512B per wave32 load).
//   Phase 2 (!TR):  1 thread = 1 channel, scalar gathers from CHW layout.
//   Phase 3: stage [c][bin] in LDS, drain with async LDS->global stores.
// ---------------------------------------------------------------------------
template <bool TR>
__global__ __launch_bounds__(256) void deform_roi_pool_kernel(
    const float* __restrict__ x,       // TR ? (B,HW,C) : (B,C,HW)
    const float* __restrict__ rois,    // (NR, 5)
    const float* __restrict__ offs_in, // (NR, 2*49)
    float* __restrict__ out)           // (NR, C, 7, 7)
{
    __shared__ int4   s_off[NSAMP_];        // 12544 B
    __shared__ float4 s_wt [NSAMP_];        // 12544 B
    __shared__ float  s_out[OUT_PER_ROI_];  // 50176 B   (total ~73.5 KB)

    const int n = blockIdx.x;
    const int t = threadIdx.x;

    // ROI parameters (uniform across block -> scalar loads)
    const float* r   = rois + n * 5;
    const int   bidx = (int)r[0];
    const float x1   = r[1] * SCALE_ - 0.5f;
    const float y1   = r[2] * SCALE_ - 0.5f;
    const float x2   = r[3] * SCALE_ - 0.5f;
    const float y2   = r[4] * SCALE_ - 0.5f;
    const float rw   = fmaxf(x2 - x1, 1.0f);
    const float rh   = fmaxf(y2 - y1, 1.0f);
    const float bw   = rw * (1.0f / P_);
    const float bh   = rh * (1.0f / P_);
    const float* op  = offs_in + n * (2 * NBIN_);

    // ---- Phase 1: sample table (shared across all 256 channels) ----
    for (int j = t; j < NSAMP_; j += 256) {
        const int bin = j >> 4;
        const int sub = j & 15;
        const int ph  = bin / 7;
        const int pw  = bin - ph * 7;
        const int sy  = sub >> 2;
        const int sx  = sub & 3;

        const float offx = GAMMA_ * rw * op[bin];
        const float offy = GAMMA_ * rh * op[NBIN_ + bin];
        const float ys = y1 + ph * bh + offy + (sy + 0.5f) * 0.25f * bh;
        const float xs = x1 + pw * bw + offx + (sx + 0.5f) * 0.25f * bw;

        const bool valid = (ys > -1.0f) && (ys < (float)H_) &&
                           (xs > -1.0f) && (xs < (float)W_);
        const float yc = fminf(fmaxf(ys, 0.0f), (float)(H_ - 1));
        const float xc = fminf(fmaxf(xs, 0.0f), (float)(W_ - 1));
        const float yf = floorf(yc);
        const float xf = floorf(xc);
        const int iy0 = (int)yf;
        const int ix0 = (int)xf;
        const int iy1 = min(iy0 + 1, H_ - 1);
        const int ix1 = min(ix0 + 1, W_ - 1);
        const float ly = yc - yf, lx = xc - xf;
        const float hy = 1.0f - ly, hx = 1.0f - lx;

        float4 w = make_float4(hy * hx, hy * lx, ly * hx, ly * lx);
        if (!valid) w = make_float4(0.f, 0.f, 0.f, 0.f);
        s_off[j] = make_int4(iy0 * W_ + ix0, iy0 * W_ + ix1,
                             iy1 * W_ + ix0, iy1 * W_ + ix1);
        s_wt[j] = w;
    }
    __syncthreads();

    // ---- Phase 2 ----
    if (TR) {
        const int grp  = t >> 6;         // 0..3 bin subgroup (2 waves each)
        const int tsub = t & 63;         // 0..63
        const int c4   = tsub * 4;       // 4 channels per thread
        const float* base = x + (size_t)bidx * HW_ * C_ + c4;

        for (int bin = grp; bin < NBIN_; bin += 4) {
            if (bin + 4 < NBIN_) {       // gfx1250 global_prefetch of next bin
                const int o0 = s_off[(bin + 4) * 16].x;
                __builtin_prefetch((const void*)(base + (size_t)o0 * C_), 0, 0);
            }
            float4 acc = make_float4(0.f, 0.f, 0.f, 0.f);
#pragma unroll 4
            for (int s = 0; s < 16; ++s) {
                const int    j = bin * 16 + s;
                const int4   o = s_off[j];   // wave-uniform: ds broadcast
                const float4 w = s_wt[j];
                const float4 v0 = *(const float4*)(base + (size_t)o.x * C_);
                const float4 v1 = *(const float4*)(base + (size_t)o.y * C_);
                const float4 v2 = *(const float4*)(base + (size_t)o.z * C_);
                const float4 v3 = *(const float4*)(base + (size_t)o.w * C_);
                fma4(acc, w.x, v0);
                fma4(acc, w.y, v1);
                fma4(acc, w.z, v2);
                fma4(acc, w.w, v3);
            }
            s_out[(c4 + 0) * NBIN_ + bin] = acc.x * (1.0f / 16.0f);
            s_out[(c4 + 1) * NBIN_ + bin] = acc.y * (1.0f / 16.0f);
            s_out[(c4 + 2) * NBIN_ + bin] = acc.z * (1.0f / 16.0f);
            s_out[(c4 + 3) * NBIN_ + bin] = acc.w * (1.0f / 16.0f);
        }
    } else {
        const int c = t;
        const float* base = x + ((size_t)bidx * C_ + c) * HW_;
        for (int bin = 0; bin < NBIN_; ++bin) {
            if (bin + 1 < NBIN_) {
                const int o0 = s_off[(bin + 1) * 16].x;
                __builtin_prefetch((const void*)(base + o0), 0, 0);
            }
            float acc = 0.0f;
#pragma unroll 4
            for (int s = 0; s < 16; ++s) {
                const int    j = bin * 16 + s;
                const int4   o = s_off[j];
                const float4 w = s_wt[j];
                acc = fmaf(w.x, base[o.x], acc);
                acc = fmaf(w.y, base[o.y], acc);
                acc = fmaf(w.z, base[o.z], acc);
                acc = fmaf(w.w, base[o.w], acc);
            }
            s_out[c * NBIN_ + bin] = acc * (1.0f / 16.0f);
        }
    }
    __syncthreads();

    // ---- Phase 3: coalesced drain of the ROI's output block ----
    float* gout = out + (size_t)n * OUT_PER_ROI_;
#if HAVE_ASYNC_LDS
    for (int i = t; i < OUT_PER_ROI_; i += 256) {
        __builtin_amdgcn_global_store_async_from_lds_b32(
            (GLOBAL_AS int*)(gout + i), (LOCAL_AS int*)(&s_out[i]), 0, 0);
    }
    __builtin_amdgcn_s_wait_asynccnt(0);
#else
    for (int i = t; i < OUT_PER_ROI_; i += 256)
        gout[i] = s_out[i];
#endif
}

// ---------------------------------------------------------------------------
extern "C" void kernel_launch(void* const* d_in, const int* in_sizes, int n_in,
                              void* d_out, int out_size, void* d_ws, size_t ws_size,
                              hipStream_t stream)
{
    (void)in_sizes; (void)n_in; (void)out_size;
    const float* x      = (const float*)d_in[0];
    const float* rois   = (const float*)d_in[1];
    const float* offset = (const float*)d_in[2];
    float*       out    = (float*)d_out;

    const size_t xT_bytes = (size_t)B_ * C_ * H_ * W_ * sizeof(float);
    if (ws_size >= xT_bytes) {
        float* xT = (float*)d_ws;
        transpose_chw_hwc<<<dim3(HW_ / 32, C_ / 32, B_), dim3(32, 8), 0, stream>>>(x, xT);
        deform_roi_pool_kernel<true><<<NR_, 256, 0, stream>>>(xT, rois, offset, out);
    } else {
        deform_roi_pool_kernel<false><<<NR_, 256, 0, stream>>>(x, rois, offset, out);
    }
}